// SequenceToSequenceLayer_47467978555533
// MI455X (gfx1250) — compile-verified
//
#include <hip/hip_runtime.h>

// Problem dims (match reference)
#define LDIM  2
#define BATCH 32
#define SEQ   512
#define FUT   64
#define IDIM  512
#define HDIM  512
#define GDIM  (4 * HDIM)

typedef __attribute__((ext_vector_type(2))) float v2f;
typedef __attribute__((ext_vector_type(8))) float v8f;

// Native fp32 WMMA: D(16x16) = A(16x4) * B(4x16) + C
__device__ __forceinline__ v8f wmma_f32(v2f a, v2f b, v8f c) {
  return __builtin_amdgcn_wmma_f32_16x16x4_f32(
      /*neg_a=*/false, a, /*neg_b=*/false, b,
      /*c_mod=*/(short)0, c, /*reuse_a=*/false, /*reuse_b=*/false);
}

// 32-bit element offset from a uniform base -> SADDR global-load addressing
__device__ __forceinline__ v2f ld2(const float* __restrict__ base, int eoff) {
  return *(const v2f*)(base + eoff);
}

__device__ __forceinline__ float sigmoidf_fast(float x) {
  return 1.0f / (1.0f + __expf(-x));
}
__device__ __forceinline__ float tanhf_fast(float x) {
  return 2.0f / (1.0f + __expf(-2.0f * x)) - 1.0f;
}

// -----------------------------------------------------------------------------
// Generic GEMM: C[(t*Bdim+b), n] = sum_k A(t,b)[k] * W[n, k] + bias0[n]+bias1[n]
// A row (t,b) fetched via element strides (handles b-major and t-major sources).
// Block = 256 threads (8 waves). Each wave: 32x64 output tile (2 A-frags x
// 4 B-frags -> 8 WMMAs / K-step, W fragments reused across both M-tiles).
// Software-pipelined: next K-step fragments prefetched before current WMMAs.
// Grid: (M/64, N/256), M = T*Bdim.
// -----------------------------------------------------------------------------
__global__ __launch_bounds__(256, 1)
void gemm_rows_bias(const float* __restrict__ A, long strideT, long strideB,
                    const float* __restrict__ W,      // [N, K] row-major
                    const float* __restrict__ bias0,  // [N] or null
                    const float* __restrict__ bias1,  // [N] or null
                    float* __restrict__ C,            // [M, N] row-major (t-major rows)
                    int Bdim, int N, int K) {
  const int lane   = threadIdx.x & 31;
  const int wave   = threadIdx.x >> 5;
  const int lane16 = lane & 15;
  const int hi     = lane >> 4;

  const int m0 = blockIdx.x * 64  + (wave >> 2) * 32;
  const int n0 = blockIdx.y * 256 + (wave & 3)  * 64;

  // A fragment rows: lanes 0-15 carry M rows with K pair {0,1}; lanes 16-31 the
  // same rows with K pair {2,3} -> one contiguous float2 per lane per K-step.
  int aoff[2];
#pragma unroll
  for (int mi = 0; mi < 2; ++mi) {
    const int row = m0 + mi * 16 + lane16;
    const int t = row / Bdim, b = row % Bdim;
    aoff[mi] = (int)((long)t * strideT + (long)b * strideB) + 2 * hi;
  }
  int woff[4];
#pragma unroll
  for (int j = 0; j < 4; ++j)
    woff[j] = (n0 + j * 16 + lane16) * K + 2 * hi;

  v8f acc[2][4];
#pragma unroll
  for (int mi = 0; mi < 2; ++mi)
#pragma unroll
    for (int j = 0; j < 4; ++j)
#pragma unroll
      for (int e = 0; e < 8; ++e) acc[mi][j][e] = 0.0f;

  // Pipeline prologue
  v2f a_c[2], b_c[4];
#pragma unroll
  for (int mi = 0; mi < 2; ++mi) a_c[mi] = ld2(A, aoff[mi]);
#pragma unroll
  for (int j = 0; j < 4; ++j)    b_c[j]  = ld2(W, woff[j]);

  for (int k = 4; k < K; k += 4) {
    v2f a_n[2], b_n[4];
#pragma unroll
    for (int mi = 0; mi < 2; ++mi) a_n[mi] = ld2(A, aoff[mi] + k);
#pragma unroll
    for (int j = 0; j < 4; ++j)    b_n[j]  = ld2(W, woff[j] + k);
#pragma unroll
    for (int mi = 0; mi < 2; ++mi)
#pragma unroll
      for (int j = 0; j < 4; ++j)
        acc[mi][j] = wmma_f32(a_c[mi], b_c[j], acc[mi][j]);
#pragma unroll
    for (int mi = 0; mi < 2; ++mi) a_c[mi] = a_n[mi];
#pragma unroll
    for (int j = 0; j < 4; ++j)    b_c[j]  = b_n[j];
  }
#pragma unroll
  for (int mi = 0; mi < 2; ++mi)
#pragma unroll
    for (int j = 0; j < 4; ++j)
      acc[mi][j] = wmma_f32(a_c[mi], b_c[j], acc[mi][j]);

  // Epilogue: C/D layout -> VGPR r holds row base+r+8*hi, col n0+j*16+lane16
#pragma unroll
  for (int j = 0; j < 4; ++j) {
    const int col = n0 + j * 16 + lane16;
    float bias = 0.0f;
    if (bias0) bias += bias0[col];
    if (bias1) bias += bias1[col];
#pragma unroll
    for (int mi = 0; mi < 2; ++mi) {
#pragma unroll
      for (int r = 0; r < 8; ++r) {
        const int orow = m0 + mi * 16 + r + 8 * hi;
        C[(size_t)orow * N + col] = acc[mi][j][r] + bias;
      }
    }
  }
}

// -----------------------------------------------------------------------------
// One LSTM timestep (one layer): gates = pre[t] + h_in @ Whh^T  (bias in pre)
// then i,f,g,o nonlinearity + cell update. Each wave owns a 16(batch) x 16(j)
// tile and computes matching i/f/g/o fragments so the elementwise stage stays
// in registers. Grid: 8 blocks x 256 threads = 64 waves = 2 m-tiles * 32 j-tiles.
// -----------------------------------------------------------------------------
__global__ __launch_bounds__(256, 1)
void lstm_step(const float* __restrict__ h_in,    // [B,H]
               float* __restrict__ h_out,         // [B,H] (ping-pong partner)
               float* __restrict__ c_st,          // [B,H] in/out (owner-updated)
               const float* __restrict__ Whh,     // [4H,H]
               const float* __restrict__ pre,     // [B,4H] slice for this t
               float* __restrict__ seq_out,       // [B,H] slice (t-major) or null
               float* __restrict__ dec_out,       // out + t*H (row stride below) or null
               long dec_stride,                   // = FUT*H
               float* __restrict__ h_fin,         // [B,H] or null
               float* __restrict__ c_fin) {       // [B,H] or null
  const int lane   = threadIdx.x & 31;
  const int wave   = threadIdx.x >> 5;
  const int w      = blockIdx.x * 8 + wave;  // 0..63
  const int m0     = (w & 1) * 16;
  const int j0     = (w >> 1) * 16;
  const int lane16 = lane & 15;
  const int hi     = lane >> 4;

  const int aoff = (m0 + lane16) * HDIM + 2 * hi;
  int woff[4];
#pragma unroll
  for (int g = 0; g < 4; ++g)
    woff[g] = (g * HDIM + j0 + lane16) * HDIM + 2 * hi;

  v8f acc[4];
#pragma unroll
  for (int g = 0; g < 4; ++g)
#pragma unroll
    for (int e = 0; e < 8; ++e) acc[g][e] = 0.0f;

  // Software-pipelined K loop: prefetch next fragments before current WMMAs.
  v2f a_c = ld2(h_in, aoff);
  v2f b_c[4];
#pragma unroll
  for (int g = 0; g < 4; ++g) b_c[g] = ld2(Whh, woff[g]);

  for (int k = 4; k < HDIM; k += 4) {
    v2f a_n = ld2(h_in, aoff + k);
    v2f b_n[4];
#pragma unroll
    for (int g = 0; g < 4; ++g) b_n[g] = ld2(Whh, woff[g] + k);
#pragma unroll
    for (int g = 0; g < 4; ++g) acc[g] = wmma_f32(a_c, b_c[g], acc[g]);
    a_c = a_n;
#pragma unroll
    for (int g = 0; g < 4; ++g) b_c[g] = b_n[g];
  }
#pragma unroll
  for (int g = 0; g < 4; ++g) acc[g] = wmma_f32(a_c, b_c[g], acc[g]);

  const int j = j0 + lane16;
#pragma unroll
  for (int r = 0; r < 8; ++r) {
    const int bidx = m0 + r + 8 * hi;
    const int poff = bidx * GDIM + j;
    const float xi = acc[0][r] + pre[poff];
    const float xf = acc[1][r] + pre[poff + HDIM];
    const float xg = acc[2][r] + pre[poff + 2 * HDIM];
    const float xo = acc[3][r] + pre[poff + 3 * HDIM];

    const float ig = sigmoidf_fast(xi);
    const float fg = sigmoidf_fast(xf);
    const float gg = tanhf_fast(xg);
    const float og = sigmoidf_fast(xo);

    const int idx = bidx * HDIM + j;
    const float c = fg * c_st[idx] + ig * gg;
    const float h = og * tanhf_fast(c);

    c_st[idx]  = c;
    h_out[idx] = h;
    if (seq_out) seq_out[idx] = h;
    if (dec_out) dec_out[(size_t)bidx * dec_stride + j] = h;
    if (h_fin) { h_fin[idx] = h; c_fin[idx] = c; }
  }
}

// -----------------------------------------------------------------------------
extern "C" void kernel_launch(void* const* d_in, const int* in_sizes, int n_in,
                              void* d_out_v, int out_size, void* d_ws, size_t ws_size,
                              hipStream_t stream) {
  (void)in_sizes; (void)n_in; (void)out_size; (void)ws_size;
  const float* x       = (const float*)d_in[0];   // [B,S,I]
  const float* fut     = (const float*)d_in[1];   // [B,F,I]
  const float* enc_h0  = (const float*)d_in[2];   // [L,B,H]
  const float* enc_c0  = (const float*)d_in[3];   // [L,B,H]
  const float* fc_w    = (const float*)d_in[4];   // [H,I]
  const float* fc_b    = (const float*)d_in[5];   // [H]
  const float* enc_wih = (const float*)d_in[6];   // [L,4H,H]
  const float* enc_whh = (const float*)d_in[7];   // [L,4H,H]
  const float* enc_bih = (const float*)d_in[8];   // [L,4H]
  const float* enc_bhh = (const float*)d_in[9];   // [L,4H]
  const float* dec_wih = (const float*)d_in[10];
  const float* dec_whh = (const float*)d_in[11];
  const float* dec_bih = (const float*)d_in[12];
  const float* dec_bhh = (const float*)d_in[13];
  float* out = (float*)d_out_v;  // dec_out[B,F,H] | hidden[L,B,H] | cell[L,B,H]

  const size_t BH = (size_t)BATCH * HDIM;
  float* ws    = (float*)d_ws;
  float* xT    = ws;                                   // [S,B,H] t-major FC output
  float* seq   = xT + (size_t)SEQ * BH;                // [S,B,H] layer-output sequence
  float* gates = seq + (size_t)SEQ * BH;               // [S,B,4H] precomputed gate inputs
  float* hbuf  = gates + (size_t)SEQ * BATCH * GDIM;   // 4*BH: (layer, parity)
  float* cbuf  = hbuf + 4 * BH;                        // 2*BH

  auto hptr = [&](int l, int p) { return hbuf + ((size_t)l * 2 + p) * BH; };

  // Initialize recurrent state from inputs (re-done on every call/replay).
  hipMemcpyAsync(hptr(0, 0), enc_h0,      BH * sizeof(float), hipMemcpyDeviceToDevice, stream);
  hipMemcpyAsync(hptr(1, 0), enc_h0 + BH, BH * sizeof(float), hipMemcpyDeviceToDevice, stream);
  hipMemcpyAsync(cbuf,       enc_c0,  2 * BH * sizeof(float), hipMemcpyDeviceToDevice, stream);

  float* hidden_out = out + (size_t)BATCH * FUT * HDIM;
  float* cell_out   = hidden_out + (size_t)LDIM * BH;

  const dim3 blk(256);

  // FC: xT[t,b,:] = x[b,t,:] @ fc_w^T + fc_b   (M=S*B, N=H, K=I)
  gemm_rows_bias<<<dim3((SEQ * BATCH) / 64, HDIM / 256), blk, 0, stream>>>(
      x, (long)IDIM, (long)SEQ * IDIM, fc_w, fc_b, nullptr, xT, BATCH, HDIM, IDIM);

  int par[2] = {0, 0};

  // ======================= encoder =======================
  for (int l = 0; l < 2; ++l) {
    const float* Ain = (l == 0) ? xT : seq;  // both t-major [T,B,H]
    gemm_rows_bias<<<dim3((SEQ * BATCH) / 64, GDIM / 256), blk, 0, stream>>>(
        Ain, (long)BATCH * HDIM, (long)HDIM,
        enc_wih + (size_t)l * GDIM * HDIM,
        enc_bih + (size_t)l * GDIM, enc_bhh + (size_t)l * GDIM,
        gates, BATCH, GDIM, HDIM);
    for (int t = 0; t < SEQ; ++t) {
      lstm_step<<<8, 256, 0, stream>>>(
          hptr(l, par[l]), hptr(l, par[l] ^ 1), cbuf + (size_t)l * BH,
          enc_whh + (size_t)l * GDIM * HDIM,
          gates + (size_t)t * BATCH * GDIM,
          (l == 0) ? (seq + (size_t)t * BH) : nullptr,  // layer 0 feeds layer 1
          nullptr, 0, nullptr, nullptr);
      par[l] ^= 1;
    }
  }

  // ======================= decoder =======================
  for (int l = 0; l < 2; ++l) {
    const float* Ain = (l == 0) ? fut : seq;
    const long sT = (l == 0) ? (long)IDIM : (long)BATCH * HDIM;
    const long sB = (l == 0) ? (long)FUT * IDIM : (long)HDIM;
    gemm_rows_bias<<<dim3((FUT * BATCH) / 64, GDIM / 256), blk, 0, stream>>>(
        Ain, sT, sB,
        dec_wih + (size_t)l * GDIM * HDIM,
        dec_bih + (size_t)l * GDIM, dec_bhh + (size_t)l * GDIM,
        gates, BATCH, GDIM, HDIM);
    for (int t = 0; t < FUT; ++t) {
      const bool last = (t == FUT - 1);
      lstm_step<<<8, 256, 0, stream>>>(
          hptr(l, par[l]), hptr(l, par[l] ^ 1), cbuf + (size_t)l * BH,
          dec_whh + (size_t)l * GDIM * HDIM,
          gates + (size_t)t * BATCH * GDIM,
          (l == 0) ? (seq + (size_t)t * BH) : nullptr,   // layer 0 feeds layer 1
          (l == 1) ? (out + (size_t)t * HDIM) : nullptr, // layer 1 -> dec_out[b,t,:]
          (long)FUT * HDIM,
          last ? (hidden_out + (size_t)l * BH) : nullptr,
          last ? (cell_out + (size_t)l * BH) : nullptr);
      par[l] ^= 1;
    }
  }
}